// GpuMeshRasterizer_1692217115427
// MI455X (gfx1250) — compile-verified
//
#include <hip/hip_runtime.h>
#include <cmath>

typedef __attribute__((ext_vector_type(2))) float v2f;
typedef __attribute__((ext_vector_type(8))) float v8f;
typedef int v4i_vs __attribute__((vector_size(16)));   // matches async builtin param

#define INFF __builtin_inff()

__device__ __forceinline__ int imin_i(int a, int b) { return a < b ? a : b; }

// ---------------------------------------------------------------------------
// Kernel A: vertex transform  vr = V @ (Ry@Rx), z += 2.
// V_WMMA_F32_16X16X4_F32: A = 16 vertices x 4 (xyz,pad), B = M padded 4x16,
// C = +2 in column 2.  One wave handles 16 vertices.
// ---------------------------------------------------------------------------
__global__ __launch_bounds__(256) void xform_kernel(
    const float* __restrict__ verts,
    const int* __restrict__ elev_p,
    const int* __restrict__ azim_p,
    float* __restrict__ vrx, float* __restrict__ vry, float* __restrict__ vrz,
    int N)
{
    const float deg = 0.017453292519943295f;
    const float e = (float)elev_p[0] * deg;
    const float a = (float)azim_p[0] * deg;
    const float ca = cosf(a), sa = sinf(a);
    const float ce = cosf(e), se = sinf(e);
    // M = rotY @ rotX:
    //  M[0] = { ca,   sa*se, sa*ce }
    //  M[1] = { 0,    ce,    -se   }
    //  M[2] = { -sa,  ca*se, ca*ce }
#if __has_builtin(__builtin_amdgcn_wmma_f32_16x16x4_f32)
    const unsigned lane = threadIdx.x & 31u;
    const unsigned wave = (blockIdx.x * blockDim.x + threadIdx.x) >> 5;
    const int base = (int)wave * 16;
    const int n  = (int)(lane & 15u);   // A: vertex row; B/D: column
    const int hi = (int)(lane >> 4);
    int vi = base + n;
    int vc = (vi < N) ? vi : 0;
    // A (16x4): lanes0-15 hold K0/K1, lanes16-31 hold K2/K3
    v2f A;
    if (!hi) { A[0] = verts[3 * vc + 0]; A[1] = verts[3 * vc + 1]; }
    else     { A[0] = verts[3 * vc + 2]; A[1] = 0.0f; }
    // B (4x16): lanes0-15 rows K0/K1, lanes16-31 rows K2/K3
    float b0, b1;
    if (!hi) {
        b0 = (n == 0) ? ca   : (n == 1) ? (sa * se) : (n == 2) ? (sa * ce) : 0.0f;
        b1 = (n == 0) ? 0.0f : (n == 1) ? ce        : (n == 2) ? (-se)     : 0.0f;
    } else {
        b0 = (n == 0) ? (-sa) : (n == 1) ? (ca * se) : (n == 2) ? (ca * ce) : 0.0f;
        b1 = 0.0f;
    }
    v2f B; B[0] = b0; B[1] = b1;
    const float cz = (n == 2) ? 2.0f : 0.0f;   // z += 2.0 via C-matrix
    v8f C;
    #pragma unroll
    for (int r = 0; r < 8; ++r) C[r] = cz;

    v8f D = __builtin_amdgcn_wmma_f32_16x16x4_f32(
        false, A, false, B, (short)0, C, false, false);

    // D layout: VGPR r, lane l -> row M = r + 8*(l>=16), col N = l&15.
    if (n < 3) {
        float* dst = (n == 0) ? vrx : (n == 1) ? vry : vrz;
        const int m0 = base + 8 * hi;
        if (base + 16 <= N) {            // common case: branch-free 8 stores
            #pragma unroll
            for (int r = 0; r < 8; ++r) dst[m0 + r] = D[r];
        } else {
            #pragma unroll
            for (int r = 0; r < 8; ++r) if (m0 + r < N) dst[m0 + r] = D[r];
        }
    }
#else
    int i = blockIdx.x * blockDim.x + threadIdx.x;
    if (i < N) {
        float x = verts[3 * i], y = verts[3 * i + 1], z = verts[3 * i + 2];
        vrx[i] = x * ca + z * (-sa);
        vry[i] = x * (sa * se) + y * ce + z * (ca * se);
        vrz[i] = x * (sa * ce) + y * (-se) + z * (ca * ce) + 2.0f;
    }
#endif
}

// ---------------------------------------------------------------------------
// Kernel B: per-triangle setup -> packed layout (13*T floats total):
//   [0, 4T)    : per-tri float4 cull  = (mnx|INF-if-invalid, mxx, mny, mxy)
//   [4T, 12T)  : per-tri 2x float4 payload = (V0x,V0y,V1x,V1y),(V2x,V2y,inv,col)
//   [12T, 13T) : per-tri z (chunk-mean depth)
// ---------------------------------------------------------------------------
__global__ __launch_bounds__(256) void tri_setup_kernel(
    const int* __restrict__ faces,
    const float* __restrict__ vrx, const float* __restrict__ vry,
    const float* __restrict__ vrz,
    float* __restrict__ tw, int T, float Sf)
{
    int t = blockIdx.x * blockDim.x + threadIdx.x;
    if (t >= T) return;
    int i0 = faces[3 * t], i1 = faces[3 * t + 1], i2 = faces[3 * t + 2];
    float p0x = vrx[i0], p0y = vry[i0], p0z = vrz[i0];
    float p1x = vrx[i1], p1y = vry[i1], p1z = vrz[i1];
    float p2x = vrx[i2], p2y = vry[i2], p2z = vrz[i2];

    // face normal -> diffuse gray
    float e1x = p1x - p0x, e1y = p1y - p0y, e1z = p1z - p0z;
    float e2x = p2x - p0x, e2y = p2y - p0y, e2z = p2z - p0z;
    float nx = e1y * e2z - e1z * e2y;
    float ny = e1z * e2x - e1x * e2z;
    float nz = e1x * e2y - e1y * e2x;
    float nrm = sqrtf(nx * nx + ny * ny + nz * nz) + 1e-8f;
    float diffuse = fminf(fmaxf(nz / nrm, 0.0f), 1.0f) * 180.0f + 75.0f;

    float zmean = (p0z + p1z + p2z) / 3.0f;

    // project + trunc (identical formula per gathered vertex)
    float half = Sf * 0.5f;
    float V0x = truncf((p0x / p0z + 1.0f) * half);
    float V0y = truncf((p0y / p0z + 1.0f) * half);
    float V1x = truncf((p1x / p1z + 1.0f) * half);
    float V1y = truncf((p1y / p1z + 1.0f) * half);
    float V2x = truncf((p2x / p2z + 1.0f) * half);
    float V2y = truncf((p2y / p2z + 1.0f) * half);

    float area = 0.5f * fabsf((V1x - V0x) * (V2y - V0y) - (V2x - V0x) * (V1y - V0y));
    bool  valid = area >= 1e-5f;
    float inv = 1.0f / (area + 1e-8f);

    float mnx = fmaxf(0.0f,      fminf(V0x, fminf(V1x, V2x)));
    float mxx = fminf(Sf - 1.0f, fmaxf(V0x, fmaxf(V1x, V2x)) + 1.0f);
    float mny = fmaxf(0.0f,      fminf(V0y, fminf(V1y, V2y)));
    float mxy = fminf(Sf - 1.0f, fmaxf(V0y, fmaxf(V1y, V2y)) + 1.0f);

    // cull float4: fold validity into x0 (invalid => +inf => always culled)
    tw[4 * t + 0] = valid ? mnx : INFF;
    tw[4 * t + 1] = mxx;
    tw[4 * t + 2] = mny;
    tw[4 * t + 3] = mxy;
    // payload 2x float4
    size_t pb = (size_t)4 * T + (size_t)8 * t;
    tw[pb + 0] = V0x; tw[pb + 1] = V0y; tw[pb + 2] = V1x; tw[pb + 3] = V1y;
    tw[pb + 4] = V2x; tw[pb + 5] = V2y; tw[pb + 6] = inv; tw[pb + 7] = diffuse;
    // z
    tw[(size_t)12 * T + t] = zmean;
}

// ---------------------------------------------------------------------------
// Kernel C: tile rasterizer.  One 256-thread block = 16x16 pixel tile.
// Whole triangle set (13*T floats = 104KB for T=2048) staged into LDS once
// via CDNA5 async-to-LDS; then wave-uniform 1x ds_load_b128 bbox cull per
// triangle, 2x b128 + 1x b32 payload only on survival.
// ---------------------------------------------------------------------------
__global__ __launch_bounds__(256) void raster_kernel(
    const float* __restrict__ tw, float* __restrict__ out, int T, int S)
{
    extern __shared__ float lds[];
    const int tid = threadIdx.x;
    const int nflt = 13 * T;

    // ---- stage into LDS ----
    {
        const int n4 = nflt >> 2;
        const float4* g = (const float4*)tw;
        float4* l = (float4*)lds;
#if __has_builtin(__builtin_amdgcn_global_load_async_to_lds_b128)
        for (int i = tid; i < n4; i += 256) {
            __builtin_amdgcn_global_load_async_to_lds_b128(
                (__attribute__((address_space(1))) v4i_vs*)(unsigned long long)(g + i),
                (__attribute__((address_space(3))) v4i_vs*)(unsigned)(unsigned long long)(l + i),
                0, 0);
        }
  #if __has_builtin(__builtin_amdgcn_s_wait_asynccnt)
        __builtin_amdgcn_s_wait_asynccnt(0);
  #else
        asm volatile("s_wait_asynccnt 0" ::: "memory");
  #endif
#else
        for (int i = tid; i < n4; i += 256) l[i] = g[i];
#endif
        for (int i = (n4 << 2) + tid; i < nflt; i += 256) lds[i] = tw[i];
    }
    __syncthreads();

    const float4* cull4 = (const float4*)lds;              // (x0,x1,y0,y1)
    const float4* pay4  = (const float4*)(lds + (size_t)4 * T);
    const float*  zA    = lds + (size_t)12 * T;

    const int lx = tid & 15, ly = tid >> 4;
    const int pxi = (int)blockIdx.x * 16 + lx;
    const int pyi = (int)blockIdx.y * 16 + ly;
    const float px = (float)pxi, py = (float)pyi;

    const float txmin = (float)((int)blockIdx.x * 16);
    const float txmax = (float)imin_i((int)blockIdx.x * 16 + 15, S - 1);
    const float tymin = (float)((int)blockIdx.y * 16);
    const float tymax = (float)imin_i((int)blockIdx.y * 16 + 15, S - 1);

    float bz = INFF, bc = 255.0f;

    for (int t = 0; t < T; ++t) {
        float4 cb = cull4[t];
        // wave-uniform tile cull (x0 = +inf for invalid -> always culled)
        bool hit = (cb.x <= txmax) & (cb.y > txmin) & (cb.z <= tymax) & (cb.w > tymin);
        if (!hit) continue;

        float4 pa = pay4[2 * t + 0];   // V0x,V0y,V1x,V1y
        float4 pb = pay4[2 * t + 1];   // V2x,V2y,inv,col
        float  z  = zA[t];

        float w0 = 0.5f * fabsf((pa.z - px) * (pb.y - py) - (pb.x - px) * (pa.w - py)) * pb.z;
        float w1 = 0.5f * fabsf((pb.x - px) * (pa.y - py) - (pa.x - px) * (pb.y - py)) * pb.z;
        float w2 = 1.0f - w0 - w1;
        bool ok = (w0 <= 1.0f) & (w1 <= 1.0f) & (w2 >= 0.0f) & (w2 <= 1.0f)
                & (px >= cb.x) & (px < cb.y) & (py >= cb.z) & (py < cb.w);
        if (ok & (z < bz)) { bz = z; bc = pb.w; }   // first strict min wins
    }

    if (pxi < S && pyi < S) {
        int o = (pyi * S + pxi) * 3;
        out[o + 0] = bc; out[o + 1] = bc; out[o + 2] = bc;
    }
}

// ---------------------------------------------------------------------------
extern "C" void kernel_launch(void* const* d_in, const int* in_sizes, int n_in,
                              void* d_out, int out_size, void* d_ws, size_t ws_size,
                              hipStream_t stream)
{
    const float* verts = (const float*)d_in[0];
    const int*   faces = (const int*)d_in[1];
    const int*   elev  = (const int*)d_in[2];
    const int*   azim  = (const int*)d_in[3];
    (void)n_in; (void)ws_size;              // d_in[4]=image_size read via out_size

    const int N = in_sizes[0] / 3;          // 10000 vertices
    const int T = in_sizes[1] / 3;          // 2048 triangles
    const int S = (int)(std::sqrt((double)(out_size / 3)) + 0.5);  // 500

    float* ws  = (float*)d_ws;
    float* vrx = ws;
    float* vry = ws + N;
    float* vrz = ws + 2 * (size_t)N;
    size_t triOff = ((size_t)(3 * N) + 63) & ~(size_t)63;  // 16B-align fields
    float* tw = ws + triOff;                               // 13*T floats

    int xThreads = ((N + 15) / 16) * 32;    // one wave per 16 vertices
    int xBlocks  = (xThreads + 255) / 256;
    xform_kernel<<<xBlocks, 256, 0, stream>>>(verts, elev, azim, vrx, vry, vrz, N);

    tri_setup_kernel<<<(T + 255) / 256, 256, 0, stream>>>(faces, vrx, vry, vrz,
                                                          tw, T, (float)S);

    dim3 grid((S + 15) / 16, (S + 15) / 16);
    size_t ldsBytes = (size_t)13 * T * sizeof(float);
    raster_kernel<<<grid, 256, ldsBytes, stream>>>(tw, (float*)d_out, T, S);
}